// GCN_77309411328660
// MI455X (gfx1250) — compile-verified
//
#include <hip/hip_runtime.h>
#include <math.h>

typedef float v2f __attribute__((ext_vector_type(2)));
typedef float v8f __attribute__((ext_vector_type(8)));

// ---------------------------------------------------------------- zero fill
__global__ __launch_bounds__(256) void zero_f32_kernel(float* __restrict__ p,
                                                       long long n) {
  long long i = (long long)blockIdx.x * blockDim.x + threadIdx.x;
  if (i < n) p[i] = 0.0f;
}

// ------------------------------------------------- edge gather + scatter-add
// WIDTH floats per node row; WIDTH/4 lanes cooperate per edge, each lane
// moves one float4 (vector gather from L2-resident features, then 4
// global_atomic_add_f32 into the aggregation buffer).
template <int WIDTH>
__global__ __launch_bounds__(256) void scatter_add_kernel(
    const float* __restrict__ feat, const int* __restrict__ src,
    const int* __restrict__ dst, float* __restrict__ agg, int nE) {
  constexpr int LPE = WIDTH / 4;  // lanes per edge
  long long g = (long long)blockIdx.x * blockDim.x + threadIdx.x;
  long long total = (long long)nE * LPE;
  if (g >= total) return;
  int e = (int)(g / LPE);
  int l = (int)(g - (long long)e * LPE);
  int s = src[e];
  int d = dst[e];
  const float4 v =
      *reinterpret_cast<const float4*>(feat + (long long)s * WIDTH + l * 4);
  float* o = agg + (long long)d * WIDTH + l * 4;
  atomicAdd(o + 0, v.x);
  atomicAdd(o + 1, v.y);
  atomicAdd(o + 2, v.z);
  atomicAdd(o + 3, v.w);
}

// ---------------------------------------------------------- fused dual GEMM
// t2 = relu(agg1 @ W1 + b1) @ W2    (fp32 WMMA 16x16x4, wave-private tiles)
//
// Block = 128 threads = 4 waves; wave w owns row tile (blockIdx.x*4 + w).
// Stage 1: one A-fragment load per k-step feeds FOUR WMMAs (the four
// 16-column slices of H = relu(agg1@W1+b1)), so the A tile is fetched from
// global exactly once per tile. H is parked in a wave-private LDS slice
// (row stride padded 64->66 floats to kill the 16-way bank conflict on the
// stage-2 column-major fragment reads). Stage 2: H(16x64) @ W2(64x16) with
// A sourced from LDS — no __syncthreads needed anywhere (intra-wave DScnt
// dependency only), and every WMMA issued is useful work.
//
// Fragment layouts per CDNA5 ISA 7.12.2 (fp32 16x4 A / 4x16 B / 16x16 C):
//   A: lane(0..15)=M,  vgpr j holds K = k0 + 2*(lane>>4) + j
//   B: lane(0..15)=N,  vgpr j holds K = k0 + 2*(lane>>4) + j
//   C/D: lanes 0-15 -> M = vgpr, lanes 16-31 -> M = vgpr + 8, N = lane&15
__global__ __launch_bounds__(128) void gcn_fused_mm_kernel(
    const float* __restrict__ agg1,  // [N,64]
    const float* __restrict__ W1,    // [64,64] row-major
    const float* __restrict__ b1,    // [64]
    const float* __restrict__ W2,    // [64,16] row-major
    float* __restrict__ t2,          // [N,16]
    int nRows) {
  constexpr int LROW = 66;  // padded LDS row stride (floats)
  __shared__ float lds[4 * 16 * LROW];

  const int lane = threadIdx.x & 31;
  const int wave = threadIdx.x >> 5;  // 0..3, one row-tile per wave
  const int half = lane >> 4;         // 0: K pair 0/1, 1: K pair 2/3
  const int l15 = lane & 15;

  const int tile = blockIdx.x * 4 + wave;
  const int rowBase = tile * 16;
  if (rowBase >= nRows) return;  // wave-uniform exit: EXEC stays all-ones

  float* myLds = lds + wave * 16 * LROW;

  // Clamp A-row reads for a ragged final tile (branchless).
  int arow = rowBase + l15;
  if (arow >= nRows) arow = nRows - 1;
  const float* aRowPtr = agg1 + (long long)arow * 64;

  // -------- stage 1: H = agg1_tile(16x64) @ W1(64x64), 4 col slices
  v8f acc0 = {}, acc1 = {}, acc2s = {}, acc3 = {};
#pragma unroll
  for (int k0 = 0; k0 < 64; k0 += 4) {
    const int kk = k0 + 2 * half;
    const float2 av = *reinterpret_cast<const float2*>(aRowPtr + kk);
    v2f a;
    a.x = av.x;
    a.y = av.y;
    const float* w1r0 = W1 + (kk + 0) * 64 + l15;
    const float* w1r1 = W1 + (kk + 1) * 64 + l15;
    v2f b0, b1v, b2v, b3;
    b0.x = w1r0[0];
    b0.y = w1r1[0];
    b1v.x = w1r0[16];
    b1v.y = w1r1[16];
    b2v.x = w1r0[32];
    b2v.y = w1r1[32];
    b3.x = w1r0[48];
    b3.y = w1r1[48];
    acc0 = __builtin_amdgcn_wmma_f32_16x16x4_f32(false, a, false, b0, (short)0,
                                                 acc0, false, false);
    acc1 = __builtin_amdgcn_wmma_f32_16x16x4_f32(false, a, false, b1v, (short)0,
                                                 acc1, false, false);
    acc2s = __builtin_amdgcn_wmma_f32_16x16x4_f32(false, a, false, b2v,
                                                  (short)0, acc2s, false,
                                                  false);
    acc3 = __builtin_amdgcn_wmma_f32_16x16x4_f32(false, a, false, b3, (short)0,
                                                 acc3, false, false);
  }

  // bias + relu -> wave-private LDS tile (16 rows x 64 cols, stride 66)
  const float bi0 = b1[0 + l15];
  const float bi1 = b1[16 + l15];
  const float bi2 = b1[32 + l15];
  const float bi3 = b1[48 + l15];
#pragma unroll
  for (int r = 0; r < 8; ++r) {
    const int row = r + 8 * half;
    float* ldr = myLds + row * LROW + l15;
    float v0 = acc0[r] + bi0;
    float v1 = acc1[r] + bi1;
    float v2 = acc2s[r] + bi2;
    float v3 = acc3[r] + bi3;
    ldr[0] = v0 > 0.0f ? v0 : 0.0f;
    ldr[16] = v1 > 0.0f ? v1 : 0.0f;
    ldr[32] = v2 > 0.0f ? v2 : 0.0f;
    ldr[48] = v3 > 0.0f ? v3 : 0.0f;
  }

  // -------- stage 2: T = H(16x64) @ W2(64x16), A from LDS (conflict-free)
  v8f acc = {};
#pragma unroll
  for (int k0 = 0; k0 < 64; k0 += 4) {
    const int kk = k0 + 2 * half;
    const float2 av =
        *reinterpret_cast<const float2*>(myLds + l15 * LROW + kk);
    v2f a;
    a.x = av.x;
    a.y = av.y;
    v2f b;
    b.x = W2[(kk + 0) * 16 + l15];
    b.y = W2[(kk + 1) * 16 + l15];
    acc = __builtin_amdgcn_wmma_f32_16x16x4_f32(false, a, false, b, (short)0,
                                                acc, false, false);
  }

#pragma unroll
  for (int r = 0; r < 8; ++r) {
    const int row = rowBase + r + 8 * half;
    if (row < nRows) t2[(long long)row * 16 + l15] = acc[r];
  }
}

// --------------------------------------------------- bias + softmax (dim=1)
__global__ __launch_bounds__(256) void softmax16_kernel(
    const float* __restrict__ agg2, const float* __restrict__ b2,
    float* __restrict__ out, int n) {
  int i = blockIdx.x * blockDim.x + threadIdx.x;
  if (i >= n) return;
  const float* p = agg2 + (long long)i * 16;
  float v[16];
  float m = -INFINITY;
#pragma unroll
  for (int j = 0; j < 16; ++j) {
    v[j] = p[j] + b2[j];
    m = fmaxf(m, v[j]);
  }
  float s = 0.0f;
#pragma unroll
  for (int j = 0; j < 16; ++j) {
    v[j] = __expf(v[j] - m);
    s += v[j];
  }
  const float inv = 1.0f / s;
  float* o = out + (long long)i * 16;
#pragma unroll
  for (int j = 0; j < 16; ++j) o[j] = v[j] * inv;
}

extern "C" void kernel_launch(void* const* d_in, const int* in_sizes, int n_in,
                              void* d_out, int out_size, void* d_ws,
                              size_t ws_size, hipStream_t stream) {
  const float* feature = (const float*)d_in[0];  // [N,64]
  const float* W1 = (const float*)d_in[1];       // [64,64]
  const float* b1 = (const float*)d_in[2];       // [64]
  const float* W2 = (const float*)d_in[3];       // [64,16]
  const float* b2 = (const float*)d_in[4];       // [16]
  const int* src = (const int*)d_in[5];          // [E]
  const int* dst = (const int*)d_in[6];          // [E]
  float* out = (float*)d_out;                    // [N,16] (flattened [N,16,1])

  const int N = in_sizes[0] / 64;
  const int E = in_sizes[5];

  // Workspace layout (fp32): agg1[N*64] | t2[N*16] | agg2[N*16]  (~38.4 MB)
  float* agg1 = (float*)d_ws;
  float* t2 = agg1 + (size_t)N * 64;
  float* agg2 = t2 + (size_t)N * 16;

  // ---- layer 1 aggregation: agg1 = segment_sum(feature[src], dst)
  {
    long long n = (long long)N * 64;
    int blocks = (int)((n + 255) / 256);
    zero_f32_kernel<<<blocks, 256, 0, stream>>>(agg1, n);
  }
  {
    long long total = (long long)E * 16;  // 16 lanes/edge * float4
    int blocks = (int)((total + 255) / 256);
    scatter_add_kernel<64>
        <<<blocks, 256, 0, stream>>>(feature, src, dst, agg1, E);
  }

  // ---- fused GEMMs: t2 = relu(agg1@W1 + b1) @ W2
  {
    int tiles = (N + 15) / 16;          // 16-row tiles
    int blocks = (tiles + 3) / 4;       // 4 wave-private tiles per block
    gcn_fused_mm_kernel<<<blocks, 128, 0, stream>>>(agg1, W1, b1, W2, t2, N);
  }

  // ---- layer 2 aggregation (reordered): agg2 = segment_sum(t2[src], dst)
  {
    long long n = (long long)N * 16;
    int blocks = (int)((n + 255) / 256);
    zero_f32_kernel<<<blocks, 256, 0, stream>>>(agg2, n);
  }
  {
    long long total = (long long)E * 4;  // 4 lanes/edge * float4
    int blocks = (int)((total + 255) / 256);
    scatter_add_kernel<16><<<blocks, 256, 0, stream>>>(t2, src, dst, agg2, E);
  }

  // ---- out = softmax(agg2 + b2, axis=1)
  {
    int blocks = (N + 255) / 256;
    softmax16_kernel<<<blocks, 256, 0, stream>>>(agg2, b2, out, N);
  }
}